// BertFinetun_80814104642308
// MI455X (gfx1250) — compile-verified
//
#include <hip/hip_runtime.h>
#include <hip/hip_bf16.h>
#include <math.h>

#define Bn 64
#define Sn 512
#define Dn 768
#define DAn 74
#define DVn 47
#define Pn 30
#define KH 96   // padded concat of 3x30 projections

typedef float v2f __attribute__((ext_vector_type(2)));
typedef float v8f __attribute__((ext_vector_type(8)));

// ---------------------------------------------------------------------------
// Kernel 1: fused modality projections -> H[b,s,0:96]
// cols 0..29 = hidden@W_t+b_t, 30..59 = audio@W_a+b_a, 60..89 = video@W_v+b_v,
// 90..95 = 0 padding so the gram kernel can run K=96 cleanly.
// ---------------------------------------------------------------------------
__global__ void proj_kernel(const float* __restrict__ hidden,
                            const float* __restrict__ audio,
                            const float* __restrict__ video,
                            const float* __restrict__ W_t, const float* __restrict__ b_t,
                            const float* __restrict__ W_a, const float* __restrict__ b_a,
                            const float* __restrict__ W_v, const float* __restrict__ b_v,
                            float* __restrict__ H) {
  int row = blockIdx.x;          // b*Sn + s
  int tid = threadIdx.x;         // 128 threads
  __shared__ float xh[Dn];
  __shared__ float xa[DAn + 2];
  __shared__ float xv[DVn + 1];
  const float* hrow = hidden + (size_t)row * Dn;
  for (int i = tid; i < Dn; i += 128) xh[i] = hrow[i];
  if (tid < DAn) xa[tid] = audio[(size_t)row * DAn + tid];
  if (tid < DVn) xv[tid] = video[(size_t)row * DVn + tid];
  __syncthreads();

  float* hout = H + (size_t)row * KH;
  if (tid < Pn) {
    float acc = b_t[tid];
    for (int d = 0; d < Dn; ++d) acc = fmaf(xh[d], W_t[d * Pn + tid], acc);
    hout[tid] = acc;
  } else if (tid < 2 * Pn) {
    int p = tid - Pn;
    float acc = b_a[p];
    for (int d = 0; d < DAn; ++d) acc = fmaf(xa[d], W_a[d * Pn + p], acc);
    hout[tid] = acc;
  } else if (tid < 3 * Pn) {
    int p = tid - 2 * Pn;
    float acc = b_v[p];
    for (int d = 0; d < DVn; ++d) acc = fmaf(xv[d], W_v[d * Pn + p], acc);
    hout[tid] = acc;
  } else if (tid < KH) {
    hout[tid] = 0.0f;          // padding
  }
}

// ---------------------------------------------------------------------------
// Kernel 2: fusion_att = H * H^T / sqrt(30), via V_WMMA_F32_16X16X4_F32.
// Each wave computes one 16x16 tile, looping K=96 in steps of 4.
// A fragment (16x4 f32): lanes 0-15 -> M=lane, v0=K0 v1=K1; lanes 16-31 ->
// M=lane-16, v0=K2 v1=K3. B (4x16) mirrors with N=lane. For a gram both
// fragments read rows of H with the identical pattern.
// C/D (16x16 f32, 8 VGPRs): vgpr j, lanes 0-15 -> M=j; lanes 16-31 -> M=j+8.
// Writes relu() to the big output; raw row-0 values to ws for the softmax.
// ---------------------------------------------------------------------------
__global__ void gram_kernel(const float* __restrict__ H,
                            float* __restrict__ att_out,
                            float* __restrict__ row0raw) {
  const int b    = blockIdx.y;
  const int wave = (int)(threadIdx.x >> 5);
  const int lane = (int)(threadIdx.x & 31);
  const int tile = blockIdx.x * 8 + wave;      // 0 .. 1023
  const int rt   = tile >> 5;                  // row tile 0..31
  const int ct   = tile & 31;                  // col tile 0..31
  const int l16  = lane & 15;
  const int hi   = lane >> 4;                  // 0: K0/K1, 1: K2/K3

  const float* Hb   = H + (size_t)b * Sn * KH;
  const float* Arow = Hb + (size_t)(rt * 16 + l16) * KH + hi * 2;
  const float* Brow = Hb + (size_t)(ct * 16 + l16) * KH + hi * 2;

  v8f c = {};
#pragma unroll
  for (int k = 0; k < KH; k += 4) {
    v2f a  = { Arow[k], Arow[k + 1] };
    v2f bb = { Brow[k], Brow[k + 1] };
    c = __builtin_amdgcn_wmma_f32_16x16x4_f32(false, a, false, bb,
                                              (short)0, c, false, false);
  }

  const float scale = 0.18257418583505536f;    // 1/sqrt(30)
  float* outb = att_out + (size_t)b * Sn * Sn;
  const int n = ct * 16 + l16;
#pragma unroll
  for (int j = 0; j < 8; ++j) {
    int m = rt * 16 + j + hi * 8;
    float v = c[j] * scale;
    outb[(size_t)m * Sn + n] = v > 0.0f ? v : 0.0f;   // relu output
    if (m == 0) row0raw[b * Sn + n] = v;              // raw logits for softmax
  }
}

// ---------------------------------------------------------------------------
// Kernel 3: per batch: softmax(row0 + mask) -> p; fd = p@hidden + hidden[0];
// h = fd@W_d + b_d; layernorm -> out1[b,:]. One 256-thread block per batch.
// ---------------------------------------------------------------------------
__global__ void head_kernel(const float* __restrict__ hidden,
                            const float* __restrict__ row0raw,
                            const float* __restrict__ mask,
                            const float* __restrict__ W_d, const float* __restrict__ b_d,
                            const float* __restrict__ ln_g, const float* __restrict__ ln_b,
                            float* __restrict__ out1) {
  const int b   = blockIdx.x;
  const int tid = (int)threadIdx.x;            // 256 threads
  __shared__ float p[Sn];
  __shared__ float fd[Dn];
  __shared__ float red[256];
  const float* hb = hidden + (size_t)b * Sn * Dn;

  // logits + max
  float lmax = -3.402823466e38f;
  for (int t = tid; t < Sn; t += 256) {
    float v = row0raw[b * Sn + t] + mask[b * Sn + t];
    p[t] = v;
    lmax = fmaxf(lmax, v);
  }
  red[tid] = lmax; __syncthreads();
  for (int s = 128; s > 0; s >>= 1) {
    if (tid < s) red[tid] = fmaxf(red[tid], red[tid + s]);
    __syncthreads();
  }
  lmax = red[0]; __syncthreads();

  // exp + sum
  float lsum = 0.0f;
  for (int t = tid; t < Sn; t += 256) {
    float e = __expf(p[t] - lmax);
    p[t] = e;
    lsum += e;
  }
  red[tid] = lsum; __syncthreads();
  for (int s = 128; s > 0; s >>= 1) {
    if (tid < s) red[tid] += red[tid + s];
    __syncthreads();
  }
  float inv = 1.0f / red[0];
  __syncthreads();

  // fusion_data row 0 (3 columns / thread)
#pragma unroll
  for (int j = 0; j < 3; ++j) {
    int d = tid + j * 256;
    float s = 0.0f;
    for (int t = 0; t < Sn; ++t) s = fmaf(p[t], hb[(size_t)t * Dn + d], s);
    fd[d] = s * inv + hb[d];
  }
  __syncthreads();

  // dense + layernorm stats
  float hv[3];
  float sum = 0.0f, sq = 0.0f;
#pragma unroll
  for (int j = 0; j < 3; ++j) {
    int e = tid + j * 256;
    float acc = b_d[e];
    for (int d = 0; d < Dn; ++d) acc = fmaf(fd[d], W_d[(size_t)d * Dn + e], acc);
    hv[j] = acc;
    sum += acc;
    sq  = fmaf(acc, acc, sq);
  }
  red[tid] = sum; __syncthreads();
  for (int s = 128; s > 0; s >>= 1) {
    if (tid < s) red[tid] += red[tid + s];
    __syncthreads();
  }
  sum = red[0]; __syncthreads();
  red[tid] = sq; __syncthreads();
  for (int s = 128; s > 0; s >>= 1) {
    if (tid < s) red[tid] += red[tid + s];
    __syncthreads();
  }
  sq = red[0]; __syncthreads();

  const float mu   = sum * (1.0f / (float)Dn);
  const float var  = sq * (1.0f / (float)Dn) - mu * mu;
  const float rstd = rsqrtf(var + 1e-12f);
#pragma unroll
  for (int j = 0; j < 3; ++j) {
    int e = tid + j * 256;
    out1[(size_t)b * Dn + e] = (hv[j] - mu) * rstd * ln_g[e] + ln_b[e];
  }
}

extern "C" void kernel_launch(void* const* d_in, const int* in_sizes, int n_in,
                              void* d_out, int out_size, void* d_ws, size_t ws_size,
                              hipStream_t stream) {
  const float* hidden = (const float*)d_in[0];
  // d_in[1] pooled_output: unused by the reference outputs
  const float* audio  = (const float*)d_in[2];
  const float* video  = (const float*)d_in[3];
  const float* mask   = (const float*)d_in[4];
  const float* W_t    = (const float*)d_in[5];
  const float* b_t    = (const float*)d_in[6];
  const float* W_a    = (const float*)d_in[7];
  const float* b_a    = (const float*)d_in[8];
  const float* W_v    = (const float*)d_in[9];
  const float* b_v    = (const float*)d_in[10];
  const float* W_d    = (const float*)d_in[11];
  const float* b_d    = (const float*)d_in[12];
  const float* ln_g   = (const float*)d_in[13];
  const float* ln_b   = (const float*)d_in[14];

  float* out1 = (float*)d_out;                   // [B, D] = h[:,0]
  float* att1 = out1 + (size_t)Bn * Dn;          // [B, S, S] = relu(fusion_att)

  float* H       = (float*)d_ws;                 // [B, S, 96]
  float* row0raw = H + (size_t)Bn * Sn * KH;     // [B, S]

  proj_kernel<<<Bn * Sn, 128, 0, stream>>>(hidden, audio, video,
                                           W_t, b_t, W_a, b_a, W_v, b_v, H);
  gram_kernel<<<dim3(128, Bn), 256, 0, stream>>>(H, att1, row0raw);
  head_kernel<<<Bn, 256, 0, stream>>>(hidden, row0raw, mask,
                                      W_d, b_d, ln_g, ln_b, out1);
}